// LARFDSSOM_64132451663918
// MI455X (gfx1250) — compile-verified
//
#include <hip/hip_runtime.h>
#include <math.h>

// ---------------- Problem constants (match reference) ----------------
#define N_NODES 4096
#define DIM     2048
#define NSAMP   64
#define A_T     0.3f
#define DSBETA  1e-4f
#define E_B     0.5f
#define E_N     (0.01f * 0.5f)
#define EPS_DS  0.01f

#define TILE_ROWS 16
#define ACT_BLOCK 128                   // 4 waves of 32, one 16-row tile per block
#define NTILES    (N_NODES / TILE_ROWS) // 256
#define UPD_BLOCK 256

typedef __attribute__((ext_vector_type(2))) float v2f;
typedef __attribute__((ext_vector_type(8))) float v8f;

// ---------------------------------------------------------------------
// Activation pass:  dist = R@x^2 - 2*(R.W)@x + (R.W^2)@1 ; relsum = R@1
// One 16-row tile per block; 4 waves each cover DIM/4 of K with
// V_WMMA_F32_16X16X4_F32, merged through LDS float atomics.
// B packs columns via per-lane constant coefficients:
//   col n==0 -> x^2, n==1 -> x, n==2 -> 1, else 0   (b = c0*x^2 + c1*x + c2)
// A/B lane layout (f32, K=4): lane L holds rows/cols (L&15); VGPR v of
// lane-group g=(L>>4) holds K = k0 + 2*g + v.  C: VGPR r, group g -> row r+8g.
// ---------------------------------------------------------------------
__global__ __launch_bounds__(ACT_BLOCK)
void som_act_kernel(const float* __restrict__ x,
                    const float* __restrict__ W,
                    const float* __restrict__ R,
                    float* __restrict__ pact,
                    int*   __restrict__ pidx)
{
    __shared__ float4 xs4[DIM / 4];            // 8 KB sample vector
    __shared__ float  red[4][TILE_ROWS];       // 0:S0 1:S1 2:S2 3:relsum
    __shared__ float  act16[TILE_ROWS];

    const int tid = threadIdx.x;
    const float4* __restrict__ x4 = (const float4*)x;
    for (int i = tid; i < DIM / 4; i += ACT_BLOCK) xs4[i] = x4[i];
    if (tid < 64) red[tid >> 4][tid & 15] = 0.0f;
    __syncthreads();

    const float* xs = (const float*)xs4;

    const int wave = tid >> 5;
    const int lane = tid & 31;
    const int m    = lane & 15;    // A-row in tile == C-column index
    const int g    = lane >> 4;    // lane group
    const int row  = blockIdx.x * TILE_ROWS + m;

    const float* __restrict__ wrow = W + (size_t)row * DIM;
    const float* __restrict__ rrow = R + (size_t)row * DIM;

    // per-lane constant B coefficients: b = c0*x^2 + c1*x + c2
    const float c0m = (m == 0) ? 1.0f : 0.0f;
    const float c1m = (m == 1) ? 1.0f : 0.0f;
    const float c2m = (m == 2) ? 1.0f : 0.0f;

    v8f c0 = {}, c1 = {}, c2 = {};
    const int kbeg = wave * (DIM / 4);
    const int kend = kbeg + (DIM / 4);

    for (int k0 = kbeg; k0 < kend; k0 += 4) {
        const int ka = k0 + 2 * g;             // this lane's two K indices: ka, ka+1
        float r0 = rrow[ka], r1 = rrow[ka + 1];
        float w0 = wrow[ka], w1 = wrow[ka + 1];

        v2f A0 = { r0,            r1            };  // R
        v2f A1 = { r0 * w0,       r1 * w1       };  // R.W
        v2f A2 = { r0 * w0 * w0,  r1 * w1 * w1  };  // R.W^2

        float xv0 = xs[ka], xv1 = xs[ka + 1];
        float b0 = fmaf(xv0, fmaf(xv0, c0m, c1m), c2m);
        float b1 = fmaf(xv1, fmaf(xv1, c0m, c1m), c2m);
        v2f Bv = { b0, b1 };

        c0 = __builtin_amdgcn_wmma_f32_16x16x4_f32(false, A0, false, Bv, (short)0, c0, false, false);
        c1 = __builtin_amdgcn_wmma_f32_16x16x4_f32(false, A1, false, Bv, (short)0, c1, false, false);
        c2 = __builtin_amdgcn_wmma_f32_16x16x4_f32(false, A2, false, Bv, (short)0, c2, false, false);
    }

    // Extract the useful C columns (col0 of c0, col1 of c1, col2 of c0/c2).
    if (m == 0) {
#pragma unroll
        for (int r = 0; r < 8; r++) atomicAdd(&red[0][8 * g + r], c0[r]);
    }
    if (m == 1) {
#pragma unroll
        for (int r = 0; r < 8; r++) atomicAdd(&red[1][8 * g + r], c1[r]);
    }
    if (m == 2) {
#pragma unroll
        for (int r = 0; r < 8; r++) {
            atomicAdd(&red[2][8 * g + r], c2[r]);
            atomicAdd(&red[3][8 * g + r], c0[r]);
        }
    }
    __syncthreads();

    if (tid < TILE_ROWS) {
        float s0  = red[0][tid];
        float s1  = red[1][tid];
        float s2  = red[2][tid];
        float rel = red[3][tid];
        float dist = s0 - 2.0f * s1 + s2;      // R@x^2 - 2(RW)@x + RW^2
        act16[tid] = rel / (rel + dist + 1e-7f);
    }
    __syncthreads();

    if (tid == 0) {                            // first-max within tile (jnp.argmax rule)
        float best = act16[0]; int bi = 0;
        for (int i = 1; i < TILE_ROWS; i++)
            if (act16[i] > best) { best = act16[i]; bi = i; }
        pact[blockIdx.x] = best;
        pidx[blockIdx.x] = blockIdx.x * TILE_ROWS + bi;
    }
}

// ---------------------------------------------------------------------
// Update pass: every block redundantly reduces the 256 tile partials to
// the global winner (first-max tie-break), then updates its own row if
// it is the winner (lr=E_B) or a neighbor (lr=E_N). All row traffic is
// float4 (b128); M_new row staged in LDS for the min/max/mean pass.
// ---------------------------------------------------------------------
__global__ __launch_bounds__(UPD_BLOCK)
void som_upd_kernel(const float* __restrict__ x,
                    const float* __restrict__ pact,
                    const int*   __restrict__ pidx,
                    const unsigned char* __restrict__ nbr,   // [N,N] bool (1B)
                    float* __restrict__ W,
                    float* __restrict__ M,
                    float* __restrict__ R)
{
    __shared__ float ract[UPD_BLOCK];
    __shared__ int   ridx[UPD_BLOCK];
    const int tid = threadIdx.x;

    ract[tid] = pact[tid];
    ridx[tid] = pidx[tid];
    __syncthreads();
#pragma unroll
    for (int s = UPD_BLOCK / 2; s > 0; s >>= 1) {
        if (tid < s) {
            float a = ract[tid], b = ract[tid + s];
            int   ia = ridx[tid], ib = ridx[tid + s];
            if (b > a || (b == a && ib < ia)) { ract[tid] = b; ridx[tid] = ib; }
        }
        __syncthreads();
    }
    const float bact   = ract[0];
    const int   winner = ridx[0];
    if (!(bact >= A_T)) return;                 // whole step is a no-op

    const int n = blockIdx.x;
    float lr;
    if (n == winner)                               lr = E_B;
    else if (nbr[(size_t)winner * N_NODES + n])    lr = E_N;
    else return;                                   // row untouched (uniform exit)

    __shared__ float4 mrow4[DIM / 4];              // 8 KB staged M_new row
    __shared__ float  rmax[UPD_BLOCK], rmin[UPD_BLOCK], rsum[UPD_BLOCK];

    float4* __restrict__ wptr = (float4*)(W + (size_t)n * DIM);
    float4* __restrict__ mptr = (float4*)(M + (size_t)n * DIM);
    float4* __restrict__ rptr = (float4*)(R + (size_t)n * DIM);
    const float4* __restrict__ x4 = (const float4*)x;

    const float a  = lr * DSBETA;
    const float oa = 1.0f - a;
    float lmax = -3.402823e38f, lmin = 3.402823e38f, lsum = 0.0f;

    const int NIT = DIM / (4 * UPD_BLOCK);         // 2 float4 iters per thread
#pragma unroll
    for (int it = 0; it < NIT; it++) {
        const int d4 = tid + it * UPD_BLOCK;
        float4 wv = wptr[d4];
        float4 xv = x4[d4];
        float4 mv = mptr[d4];
        float4 df, wn, mn;
        df.x = xv.x - wv.x;  df.y = xv.y - wv.y;  df.z = xv.z - wv.z;  df.w = xv.w - wv.w;
        wn.x = fmaf(lr, df.x, wv.x);  wn.y = fmaf(lr, df.y, wv.y);
        wn.z = fmaf(lr, df.z, wv.z);  wn.w = fmaf(lr, df.w, wv.w);
        mn.x = fmaf(a, fabsf(df.x), oa * mv.x);  mn.y = fmaf(a, fabsf(df.y), oa * mv.y);
        mn.z = fmaf(a, fabsf(df.z), oa * mv.z);  mn.w = fmaf(a, fabsf(df.w), oa * mv.w);
        wptr[d4]  = wn;                            // W update (b128 store)
        mrow4[d4] = mn;                            // stage M_new in LDS
        lmax = fmaxf(lmax, fmaxf(fmaxf(mn.x, mn.y), fmaxf(mn.z, mn.w)));
        lmin = fminf(lmin, fminf(fminf(mn.x, mn.y), fminf(mn.z, mn.w)));
        lsum += (mn.x + mn.y) + (mn.z + mn.w);
    }
    rmax[tid] = lmax; rmin[tid] = lmin; rsum[tid] = lsum;
    __syncthreads();
#pragma unroll
    for (int s = UPD_BLOCK / 2; s > 0; s >>= 1) {
        if (tid < s) {
            rmax[tid] = fmaxf(rmax[tid], rmax[tid + s]);
            rmin[tid] = fminf(rmin[tid], rmin[tid + s]);
            rsum[tid] += rsum[tid + s];
        }
        __syncthreads();
    }
    const float mx    = rmax[0];
    const float mnv   = rmin[0];
    const float av    = rsum[0] / (float)DIM;
    const float denom = EPS_DS * (mx - mnv);
    const bool  dz    = (denom == 0.0f);
    const float inv   = dz ? 0.0f : 1.0f / denom;

#pragma unroll
    for (int it = 0; it < NIT; it++) {
        const int d4 = tid + it * UPD_BLOCK;
        float4 mv = mrow4[d4];
        mptr[d4]  = mv;                            // M writeback (b128 store)
        float4 rv;
        {
            float t;
            t = fminf(fmaxf((mv.x - av) * inv, -80.0f), 80.0f);
            rv.x = dz ? 1.0f : 1.0f / (1.0f + expf(t));
            t = fminf(fmaxf((mv.y - av) * inv, -80.0f), 80.0f);
            rv.y = dz ? 1.0f : 1.0f / (1.0f + expf(t));
            t = fminf(fmaxf((mv.z - av) * inv, -80.0f), 80.0f);
            rv.z = dz ? 1.0f : 1.0f / (1.0f + expf(t));
            t = fminf(fmaxf((mv.w - av) * inv, -80.0f), 80.0f);
            rv.w = dz ? 1.0f : 1.0f / (1.0f + expf(t));
        }
        rptr[d4] = rv;                             // R writeback (b128 store)
    }
}

// ---------------------------------------------------------------------
// Host side: working W lives in d_out; M/R working copies + argmax
// partials in d_ws (needs 2*N*D*4 B + 2 KB = 64 MB + eps).
// 64 sequential (act, upd) pairs on the stream; graph-capture safe.
// ---------------------------------------------------------------------
extern "C" void kernel_launch(void* const* d_in, const int* in_sizes, int n_in,
                              void* d_out, int out_size, void* d_ws, size_t ws_size,
                              hipStream_t stream) {
    const float*         X   = (const float*)d_in[0];          // [B, D]
    const float*         W0  = (const float*)d_in[1];          // [N, D]
    const float*         M0  = (const float*)d_in[2];          // [N, D]
    const float*         R0  = (const float*)d_in[3];          // [N, D]
    const unsigned char* NBR = (const unsigned char*)d_in[5];  // [N, N] bool

    float* Wd   = (float*)d_out;                    // working weights == output
    float* Mw   = (float*)d_ws;
    float* Rw   = Mw + (size_t)N_NODES * DIM;
    float* pact = Rw + (size_t)N_NODES * DIM;
    int*   pidx = (int*)(pact + NTILES);

    const size_t bytes = (size_t)N_NODES * DIM * sizeof(float);
    hipMemcpyAsync(Wd, W0, bytes, hipMemcpyDeviceToDevice, stream);
    hipMemcpyAsync(Mw, M0, bytes, hipMemcpyDeviceToDevice, stream);
    hipMemcpyAsync(Rw, R0, bytes, hipMemcpyDeviceToDevice, stream);

    for (int b = 0; b < NSAMP; b++) {
        const float* xb = X + (size_t)b * DIM;
        som_act_kernel<<<NTILES, ACT_BLOCK, 0, stream>>>(xb, Wd, Rw, pact, pidx);
        som_upd_kernel<<<N_NODES, UPD_BLOCK, 0, stream>>>(xb, pact, pidx, NBR, Wd, Mw, Rw);
    }
}